// ProbSparseAttention_39101382263280
// MI455X (gfx1250) — compile-verified
//
#include <hip/hip_runtime.h>
#include <hip/hip_bf16.h>
#include <math.h>

// Problem constants (from the reference)
constexpr int B_  = 4;
constexpr int L_  = 2048;
constexpr int H_  = 8;
constexpr int D_  = 64;
constexpr int S_  = 40;   // sampled keys per query
constexpr int U_  = 40;   // top-k queries kept
constexpr int UP_ = 48;   // padded to 3 WMMA 16-row tiles
constexpr int BH_ = B_ * H_;
constexpr int NC_ = 16;   // cumsum chunks
constexpr int CH_ = L_ / NC_;

typedef __attribute__((ext_vector_type(2))) float v2f;
typedef __attribute__((ext_vector_type(8))) float v8f;

// ---------------------------------------------------------------------------
// K1: sparsity measure M[b,h,l] = max_s(q.k_s) - mean over L of sampled dots
// One thread per (b,h,l). K slice is L2-resident (512KB per bh).
// ---------------------------------------------------------------------------
__global__ void measure_kernel(const float* __restrict__ Q,
                               const float* __restrict__ K,
                               const int* __restrict__ idx,
                               float* __restrict__ M) {
  int t = blockIdx.x * blockDim.x + threadIdx.x;
  if (t >= BH_ * L_) return;
  int l  = t % L_;
  int bh = t / L_;
  int h  = bh % H_;
  int b  = bh / H_;
  const float4* q = (const float4*)(Q + (((size_t)b * L_ + l) * H_ + h) * D_);
  float4 qv[16];
#pragma unroll
  for (int i = 0; i < 16; ++i) qv[i] = q[i];
  float mx = -INFINITY, sm = 0.f;
  for (int s = 0; s < S_; ++s) {
    int ks = idx[l * S_ + s];
    const float4* kp = (const float4*)(K + (((size_t)b * L_ + ks) * H_ + h) * D_);
    float d = 0.f;
#pragma unroll
    for (int i = 0; i < 16; ++i) {
      float4 kv = kp[i];
      d += qv[i].x * kv.x + qv[i].y * kv.y + qv[i].z * kv.z + qv[i].w * kv.w;
    }
    mx = fmaxf(mx, d);
    sm += d;
  }
  M[t] = mx - sm * (1.0f / (float)L_);
}

// ---------------------------------------------------------------------------
// K2: per-(b,h) top-40 of M via 40 iterative argmax reductions (tie -> lowest
// index, matching lax.top_k), then gather Q_reduce[48][64] (pad rows = 0).
// ---------------------------------------------------------------------------
__global__ void topk_kernel(const float* __restrict__ M,
                            const float* __restrict__ Q,
                            int* __restrict__ Mtop,
                            float* __restrict__ Qr) {
  __shared__ float vals[L_];
  __shared__ float rv[256];
  __shared__ int   ri[256];
  __shared__ int   top[UP_];
  int bh = blockIdx.x;
  int h = bh % H_, b = bh / H_;
  int tid = threadIdx.x;
  for (int i = tid; i < L_; i += 256) vals[i] = M[(size_t)bh * L_ + i];
  if (tid < UP_) top[tid] = -1;
  __syncthreads();
  for (int it = 0; it < U_; ++it) {
    float best = -INFINITY; int bidx = L_;
    for (int i = tid; i < L_; i += 256) {
      float v = vals[i];
      if (v > best) { best = v; bidx = i; }
    }
    rv[tid] = best; ri[tid] = bidx;
    __syncthreads();
    for (int off = 128; off > 0; off >>= 1) {
      if (tid < off) {
        float v2 = rv[tid + off]; int i2 = ri[tid + off];
        if (v2 > rv[tid] || (v2 == rv[tid] && i2 < ri[tid])) { rv[tid] = v2; ri[tid] = i2; }
      }
      __syncthreads();
    }
    if (tid == 0) { top[it] = ri[0]; vals[ri[0]] = -INFINITY; }
    __syncthreads();
  }
  if (tid < UP_) Mtop[bh * UP_ + tid] = top[tid];
  for (int i = tid; i < UP_ * D_; i += 256) {
    int r = i / D_, d = i % D_;
    int src = top[r];
    float v = 0.f;
    if (src >= 0) v = Q[(((size_t)b * L_ + src) * H_ + h) * D_ + d];
    Qr[(size_t)bh * UP_ * D_ + i] = v;
  }
}

// ---------------------------------------------------------------------------
// K3: scores[bh][48][2048] = (Qr @ K^T) * 1/sqrt(D) via v_wmma_f32_16x16x4_f32
// One wave per block; block handles one 16-row m-tile x 8 n-tiles.
// A frag (lane): {Qr[m][4k+2*half], Qr[m][4k+2*half+1]},  m = lane%16
// B frag (lane): {K[n][4k+2*half],  K[n][4k+2*half+1]},   n = n0 + lane%16
// ---------------------------------------------------------------------------
__global__ void scores_kernel(const float* __restrict__ Qr,
                              const float* __restrict__ Kmat,
                              float* __restrict__ Sc) {
  int blk = blockIdx.x;
  int ntb = blk % 16; blk /= 16;   // 16 blocks of 8 n-tiles = 128 n-tiles
  int mt  = blk % 3;  blk /= 3;
  int bh  = blk;
  int h = bh % H_, b = bh / H_;
  int lane = threadIdx.x;
  int half = lane >> 4;
  int lm   = lane & 15;

  const float* qrow = Qr + (size_t)bh * UP_ * D_ + (size_t)(mt * 16 + lm) * D_ + 2 * half;
  v2f afrag[16];
#pragma unroll
  for (int k = 0; k < 16; ++k) { afrag[k].x = qrow[4 * k]; afrag[k].y = qrow[4 * k + 1]; }

  for (int j = 0; j < 8; ++j) {
    int n0 = (ntb * 8 + j) * 16;
    const float* krow = Kmat + (((size_t)b * L_ + n0 + lm) * H_ + h) * D_ + 2 * half;
    v8f c = {};
#pragma unroll
    for (int k = 0; k < 16; ++k) {
      v2f bfrag; bfrag.x = krow[4 * k]; bfrag.y = krow[4 * k + 1];
      c = __builtin_amdgcn_wmma_f32_16x16x4_f32(false, afrag[k], false, bfrag,
                                                (short)0, c, false, false);
    }
    float* srow = Sc + ((size_t)bh * UP_ + mt * 16) * L_ + n0;
#pragma unroll
    for (int i = 0; i < 8; ++i) {
      int m = i + 8 * half;
      srow[(size_t)m * L_ + lm] = c[i] * 0.125f;   // 1/sqrt(64)
    }
  }
}

// ---------------------------------------------------------------------------
// K4: row softmax over the 40 real rows (pad rows stay zero -> discarded).
// ---------------------------------------------------------------------------
__global__ void softmax_kernel(float* __restrict__ Sc) {
  __shared__ float red[256];
  int r = blockIdx.x;
  int m = r % U_, bh = r / U_;
  float* row = Sc + ((size_t)bh * UP_ + m) * L_;
  int tid = threadIdx.x;
  float mx = -INFINITY;
  for (int i = tid; i < L_; i += 256) mx = fmaxf(mx, row[i]);
  red[tid] = mx; __syncthreads();
  for (int off = 128; off > 0; off >>= 1) {
    if (tid < off) red[tid] = fmaxf(red[tid], red[tid + off]);
    __syncthreads();
  }
  mx = red[0]; __syncthreads();
  float sm = 0.f;
  for (int i = tid; i < L_; i += 256) { float e = __expf(row[i] - mx); row[i] = e; sm += e; }
  red[tid] = sm; __syncthreads();
  for (int off = 128; off > 0; off >>= 1) {
    if (tid < off) red[tid] += red[tid + off];
    __syncthreads();
  }
  float inv = 1.0f / red[0];
  for (int i = tid; i < L_; i += 256) row[i] *= inv;
}

// ---------------------------------------------------------------------------
// Cumsum of V along L into out (both laid out (B,L,H,D)): blocked 2-phase scan.
// ---------------------------------------------------------------------------
__global__ void csum_partial(const float* __restrict__ V, float* __restrict__ csums) {
  int blk = blockIdx.x;
  int c = blk % NC_, bh = blk / NC_;
  int h = bh % H_, b = bh / H_;
  int d = threadIdx.x;
  float acc = 0.f;
  int l0 = c * CH_;
  for (int l = l0; l < l0 + CH_; ++l)
    acc += V[(((size_t)b * L_ + l) * H_ + h) * D_ + d];
  csums[((size_t)bh * NC_ + c) * D_ + d] = acc;
}

__global__ void csum_scan(float* __restrict__ csums) {
  int bh = blockIdx.x;
  int d = threadIdx.x;
  float run = 0.f;
  for (int c = 0; c < NC_; ++c) {
    size_t i = ((size_t)bh * NC_ + c) * D_ + d;
    float v = csums[i];
    csums[i] = run;
    run += v;
  }
}

__global__ void csum_write(const float* __restrict__ V, const float* __restrict__ csums,
                           float* __restrict__ out) {
  int blk = blockIdx.x;
  int c = blk % NC_, bh = blk / NC_;
  int h = bh % H_, b = bh / H_;
  int d = threadIdx.x;
  float run = csums[((size_t)bh * NC_ + c) * D_ + d];
  int l0 = c * CH_;
  for (int l = l0; l < l0 + CH_; ++l) {
    size_t i = (((size_t)b * L_ + l) * H_ + h) * D_ + d;
    run += V[i];
    out[i] = run;
  }
}

// ---------------------------------------------------------------------------
// K5: upd = attn(16x2048) @ V(2048x64) via chained wmma (K=4 steps, 4 n-tiles),
// then scatter the 40 real rows over the cumsum output.
// B frag (lane): {V[4kk+2*half][n0+lm], V[4kk+2*half+1][n0+lm]}
// ---------------------------------------------------------------------------
__global__ void update_kernel(const float* __restrict__ Sc,
                              const float* __restrict__ V,
                              const int* __restrict__ Mtop,
                              float* __restrict__ out) {
  int blk = blockIdx.x;
  int mt = blk % 3, bh = blk / 3;
  int h = bh % H_, b = bh / H_;
  int lane = threadIdx.x;
  int half = lane >> 4;
  int lm   = lane & 15;

  const float* arow = Sc + ((size_t)bh * UP_ + mt * 16 + lm) * L_ + 2 * half;
  v8f c0 = {}, c1 = {}, c2 = {}, c3 = {};
  for (int kk = 0; kk < L_ / 4; ++kk) {
    v2f a; a.x = arow[4 * kk]; a.y = arow[4 * kk + 1];
    int krow = 4 * kk + 2 * half;
    const float* vp0 = V + (((size_t)b * L_ + krow) * H_ + h) * D_;
    const float* vp1 = vp0 + (size_t)H_ * D_;   // next L row
    v2f bf;
    bf.x = vp0[lm +  0]; bf.y = vp1[lm +  0];
    c0 = __builtin_amdgcn_wmma_f32_16x16x4_f32(false, a, false, bf, (short)0, c0, false, false);
    bf.x = vp0[lm + 16]; bf.y = vp1[lm + 16];
    c1 = __builtin_amdgcn_wmma_f32_16x16x4_f32(false, a, false, bf, (short)0, c1, false, false);
    bf.x = vp0[lm + 32]; bf.y = vp1[lm + 32];
    c2 = __builtin_amdgcn_wmma_f32_16x16x4_f32(false, a, false, bf, (short)0, c2, false, false);
    bf.x = vp0[lm + 48]; bf.y = vp1[lm + 48];
    c3 = __builtin_amdgcn_wmma_f32_16x16x4_f32(false, a, false, bf, (short)0, c3, false, false);
  }
#pragma unroll
  for (int i = 0; i < 8; ++i) {
    int m = mt * 16 + i + 8 * half;
    if (m < U_) {
      int l = Mtop[bh * UP_ + m];
      float* orow = out + (((size_t)b * L_ + l) * H_ + h) * D_;
      orow[lm +  0] = c0[i];
      orow[lm + 16] = c1[i];
      orow[lm + 32] = c2[i];
      orow[lm + 48] = c3[i];
    }
  }
}

// ---------------------------------------------------------------------------
extern "C" void kernel_launch(void* const* d_in, const int* in_sizes, int n_in,
                              void* d_out, int out_size, void* d_ws, size_t ws_size,
                              hipStream_t stream) {
  (void)in_sizes; (void)n_in; (void)out_size; (void)ws_size;
  const float* Q   = (const float*)d_in[0];
  const float* K   = (const float*)d_in[1];
  const float* V   = (const float*)d_in[2];
  const int*   idx = (const int*)d_in[3];
  float* out = (float*)d_out;

  // workspace carve-up (offsets are 256B-aligned)
  char* ws = (char*)d_ws;
  float* Mvals = (float*)(ws);                                   // BH*L floats      (256 KB)
  int*   Mtop  = (int*)  (ws + 262144);                          // BH*48 ints       (6 KB)
  float* Qr    = (float*)(ws + 268288);                          // BH*48*64 floats  (384 KB)
  float* csums = (float*)(ws + 661504);                          // BH*16*64 floats  (128 KB)
  float* Sc    = (float*)(ws + 792576);                          // BH*48*2048 floats (12 MB)

  measure_kernel<<<dim3((BH_ * L_ + 255) / 256), dim3(256), 0, stream>>>(Q, K, idx, Mvals);
  topk_kernel<<<dim3(BH_), dim3(256), 0, stream>>>(Mvals, Q, Mtop, Qr);
  scores_kernel<<<dim3(BH_ * 3 * 16), dim3(32), 0, stream>>>(Qr, K, Sc);
  softmax_kernel<<<dim3(BH_ * U_), dim3(256), 0, stream>>>(Sc);
  csum_partial<<<dim3(BH_ * NC_), dim3(D_), 0, stream>>>(V, csums);
  csum_scan<<<dim3(BH_), dim3(D_), 0, stream>>>(csums);
  csum_write<<<dim3(BH_ * NC_), dim3(D_), 0, stream>>>(V, csums, out);
  update_kernel<<<dim3(BH_ * 3), dim3(32), 0, stream>>>(Sc, V, Mtop, out);
}